// CondInvertible1x1Conv_42923903156930
// MI455X (gfx1250) — compile-verified
//
#include <hip/hip_runtime.h>
#include <math.h>

typedef float v2f __attribute__((ext_vector_type(2)));
typedef float v8f __attribute__((ext_vector_type(8)));

#define BATCH   64
#define CCH     16
#define HW      (128 * 128)          // 16384
#define COND_SZ 128
#define FC      512
#define Z_ELEMS (BATCH * CCH * HW)   // 16,777,216

// ---------------------------------------------------------------------------
// Kernel 1: cond MLP -> per-sample 16x16 weight matrix (tanh output)
// One block (256 threads) per sample. All tiny; VALU fp32 is fine.
// ---------------------------------------------------------------------------
__global__ void mlp_weight_kernel(const float* __restrict__ cond,
                                  const float* __restrict__ W1, const float* __restrict__ b1,
                                  const float* __restrict__ W2, const float* __restrict__ b2,
                                  const float* __restrict__ W3, const float* __restrict__ b3,
                                  float* __restrict__ wgt /* [B,16,16] */) {
    __shared__ float s_cond[COND_SZ];
    __shared__ float s_h1[FC];
    __shared__ float s_h2[FC];

    const int b   = blockIdx.x;
    const int tid = threadIdx.x;

    if (tid < COND_SZ) s_cond[tid] = cond[b * COND_SZ + tid];
    __syncthreads();

    // hid1 = relu(cond @ W1^T + b1): 512 outputs, dot length 128
    #pragma unroll
    for (int r = 0; r < 2; ++r) {
        const int j = tid + r * 256;
        const float* wrow = W1 + (size_t)j * COND_SZ;
        float acc = b1[j];
        for (int k = 0; k < COND_SZ; ++k) acc = fmaf(s_cond[k], wrow[k], acc);
        s_h1[j] = fmaxf(acc, 0.0f);
    }
    __syncthreads();

    // hid2 = relu(hid1 @ W2^T + b2): 512 outputs, dot length 512
    #pragma unroll
    for (int r = 0; r < 2; ++r) {
        const int j = tid + r * 256;
        const float* wrow = W2 + (size_t)j * FC;
        float acc = b2[j];
        for (int k = 0; k < FC; ++k) acc = fmaf(s_h1[k], wrow[k], acc);
        s_h2[j] = fmaxf(acc, 0.0f);
    }
    __syncthreads();

    // weight = tanh(hid2 @ W3^T + b3): 256 outputs, dot length 512
    if (tid < CCH * CCH) {
        const float* wrow = W3 + (size_t)tid * FC;
        float acc = b3[tid];
        for (int k = 0; k < FC; ++k) acc = fmaf(s_h2[k], wrow[k], acc);
        wgt[b * (CCH * CCH) + tid] = tanhf(acc);
    }
}

// ---------------------------------------------------------------------------
// Kernel 2: log|det| of each 16x16 weight via LU w/ partial pivoting.
// One thread per sample (64 threads total) — trivial work.
// ---------------------------------------------------------------------------
__global__ void slogdet_kernel(const float* __restrict__ wgt,
                               float* __restrict__ dlogdet) {
    const int b = threadIdx.x;
    if (b >= BATCH) return;

    float a[CCH][CCH];
    const float* src = wgt + b * (CCH * CCH);
    for (int i = 0; i < CCH; ++i)
        for (int j = 0; j < CCH; ++j)
            a[i][j] = src[i * CCH + j];

    float ld = 0.0f;
    for (int k = 0; k < CCH; ++k) {
        int p = k;
        float amax = fabsf(a[k][k]);
        for (int i = k + 1; i < CCH; ++i) {
            float v = fabsf(a[i][k]);
            if (v > amax) { amax = v; p = i; }
        }
        if (p != k) {
            for (int j = 0; j < CCH; ++j) {
                float t = a[k][j]; a[k][j] = a[p][j]; a[p][j] = t;
            }
        }
        const float piv = a[k][k];
        ld += logf(fabsf(piv));
        const float inv = 1.0f / piv;
        for (int i = k + 1; i < CCH; ++i) {
            const float f = a[i][k] * inv;
            for (int j = k + 1; j < CCH; ++j) a[i][j] = fmaf(-f, a[k][j], a[i][j]);
        }
    }
    dlogdet[b] = ld * (float)HW;
}

// ---------------------------------------------------------------------------
// Kernel 3: z[b] = weight[b] (16x16) @ x[b] (16 x 16384) via fp32 WMMA.
// V_WMMA_F32_16X16X4_F32 chained 4x over K=16. One 16x16 output tile per
// wave per iteration. Grid: (64 samples, 8) x 8 waves -> 16 tiles/wave.
//
// VGPR layouts (ISA 7.12.2, 32-bit elements):
//   A 16x4 : lane<16 -> M=lane,   (K=k0,k0+1) ; lane>=16 -> M=lane-16, (K=k0+2,k0+3)
//   B 4x16 : v.x row = k0 + 2*laneHi ; v.y row = k0+1 + 2*laneHi ; N = lane&15
//   C 16x16: vgpr r -> M = r + 8*laneHi ; N = lane&15
// ---------------------------------------------------------------------------
__global__ void einsum_wmma_kernel(const float* __restrict__ wgt,
                                   const float* __restrict__ x,
                                   float* __restrict__ z) {
    const int b      = blockIdx.x;
    const int wave   = threadIdx.x >> 5;   // 0..7
    const int lane   = threadIdx.x & 31;
    const int laneHi = lane >> 4;          // 0 or 1
    const int lane16 = lane & 15;

    const float* Wb = wgt + b * (CCH * CCH);
    const float* Xb = x + (size_t)b * CCH * HW;
    float*       Zb = z + (size_t)b * CCH * HW;

    // A-matrix: full 16x16 weight, 4 K-slices of 16x4, resident in 8 VGPRs.
    v2f amat[4];
    #pragma unroll
    for (int s = 0; s < 4; ++s) {
        const int col = 4 * s + 2 * laneHi;          // even -> 8B aligned
        amat[s] = *(const v2f*)(Wb + lane16 * CCH + col);
    }

    const int tilesPerWave = (HW / 16) / (gridDim.y * 8);       // 16
    const int tbase = (blockIdx.y * 8 + wave) * tilesPerWave;

    for (int t = 0; t < tilesPerWave; ++t) {
        const int n0 = (tbase + t) * 16;

        // B-matrix K-slices: 4x16 each (2 VGPRs)
        v2f bmat[4];
        #pragma unroll
        for (int s = 0; s < 4; ++s) {
            const int krow = 4 * s + 2 * laneHi;
            bmat[s].x = Xb[(size_t)(krow)     * HW + n0 + lane16];
            bmat[s].y = Xb[(size_t)(krow + 1) * HW + n0 + lane16];
        }

        v8f acc = {0.f, 0.f, 0.f, 0.f, 0.f, 0.f, 0.f, 0.f};
        #pragma unroll
        for (int s = 0; s < 4; ++s) {
            acc = __builtin_amdgcn_wmma_f32_16x16x4_f32(
                /*neg_a=*/false, amat[s], /*neg_b=*/false, bmat[s],
                /*c_mod=*/(short)0, acc, /*reuse_a=*/false, /*reuse_b=*/false);
        }

        #pragma unroll
        for (int r = 0; r < 8; ++r) {
            const int m = r + 8 * laneHi;
            Zb[(size_t)m * HW + n0 + lane16] = acc[r];
        }
    }
}

// ---------------------------------------------------------------------------
extern "C" void kernel_launch(void* const* d_in, const int* in_sizes, int n_in,
                              void* d_out, int out_size, void* d_ws, size_t ws_size,
                              hipStream_t stream) {
    const float* x    = (const float*)d_in[0];
    const float* cond = (const float*)d_in[1];
    const float* W1   = (const float*)d_in[2];
    const float* b1   = (const float*)d_in[3];
    const float* W2   = (const float*)d_in[4];
    const float* b2   = (const float*)d_in[5];
    const float* W3   = (const float*)d_in[6];
    const float* b3   = (const float*)d_in[7];

    float* out = (float*)d_out;                 // [z (B*C*HW) | dlogdet (B)]
    float* wgt = (float*)d_ws;                  // B*256 floats = 64 KB scratch

    mlp_weight_kernel<<<BATCH, 256, 0, stream>>>(cond, W1, b1, W2, b2, W3, b3, wgt);
    slogdet_kernel<<<1, 64, 0, stream>>>(wgt, out + Z_ELEMS);
    einsum_wmma_kernel<<<dim3(BATCH, 8), 256, 0, stream>>>(wgt, x, out);
}